// tidigitsGRUPT_47021301956609
// MI455X (gfx1250) — compile-verified
//
#include <hip/hip_runtime.h>
#include <hip/hip_bf16.h>

typedef __attribute__((ext_vector_type(2))) float v2f;
typedef __attribute__((ext_vector_type(8))) float v8f;

#define DEVINL __device__ __forceinline__

namespace {
constexpr int kB  = 1024;
constexpr int kT  = 254;
constexpr int kF  = 39;
constexpr int kH  = 200;
constexpr int kC  = 10;
constexpr int kTF = kT * kF;   // 9906
constexpr int kK1 = kT * kH;   // 50800 (FC1 reduction length)
constexpr int kHS = 202;       // padded LDS row stride (202 mod 64 = 10 -> near conflict-free)
}

// D = A(16x4,f32) * B(4x16,f32) + C(16x16,f32), full fp32 matrix op on CDNA5
DEVINL v8f wmma_f32_k4(v2f a, v2f b, v8f c) {
  return __builtin_amdgcn_wmma_f32_16x16x4_f32(false, a, false, b, (short)0, c, false, false);
}
DEVINL float sigmoid_f(float x) { return 1.0f / (1.0f + __expf(-x)); }
DEVINL float tanh_f(float x)    { return 1.0f - 2.0f / (__expf(2.0f * x) + 1.0f); }

// One block owns 16 batch rows for the whole sequence.
//  - h kept in LDS (16 x 200 f32, padded stride)
//  - per step: gi GEMM (K=39->40), gh GEMM (K=200), in-register gates, FC1 partial (K=200)
//  - FC1 accumulators persist in VGPRs across all 254 steps; stored once at the end.
//
// Guarding strategy (keeps inner loops branch-free, EXEC all-1s for WMMA):
//  - column guard (jcol >= 200): B row index clamped to row 0 (finite data); the
//    resulting garbage D columns are never read (gate math + stores are guarded).
//  - K-tail guard (k=39): only A.y is zeroed (select on clamped unconditional load);
//    a zero A element annihilates the whole k-slice, so B tail loads are just clamped.
__global__ __launch_bounds__(256, 1)
void gru_fc1_fused(const float* __restrict__ x,
                   const float* __restrict__ h0,
                   const float* __restrict__ w_ih,
                   const float* __restrict__ w_hh,
                   const float* __restrict__ b_ih,
                   const float* __restrict__ b_hh,
                   const float* __restrict__ fc1_w,
                   float* __restrict__ hidden) {
  __shared__ float hbuf[16 * kHS];

  const int tid  = threadIdx.x;
  const int wave = tid >> 5;          // 8 waves (wave32)
  const int lane = tid & 31;
  const int n16  = lane & 15;         // column within 16-wide tile / A-matrix row
  const int hi   = lane >> 4;         // half-wave select (K pair / M+8 in D layout)
  const int b0   = blockIdx.x * 16;   // batch tile base

  // stage h0 tile into LDS
  for (int idx = tid; idx < 16 * kH; idx += 256) {
    const int r = idx / kH;
    const int c = idx - r * kH;
    hbuf[r * kHS + c] = h0[(b0 + r) * kH + c];
  }
  __syncthreads();

  // each wave owns hidden-column tiles jt = wave and wave+8 (13 valid tiles cover H=200)
  int jcol[2], jc[2];
  bool vj[2];
  float br[2], bz[2], bin[2], bhn[2];
  const float* wiR[2]; const float* wiZ[2]; const float* wiN[2];
  const float* whR[2]; const float* whZ[2]; const float* whN[2];
  const float* fwb[2];
#pragma unroll
  for (int g = 0; g < 2; ++g) {
    const int jt = wave + 8 * g;
    jcol[g] = jt * 16 + n16;
    vj[g]   = (jcol[g] < kH);
    jc[g]   = vj[g] ? jcol[g] : 0;    // clamped: loads stay in-bounds & finite
    br[g]  = b_ih[jc[g]] + b_hh[jc[g]];
    bz[g]  = b_ih[kH + jc[g]] + b_hh[kH + jc[g]];
    bin[g] = b_ih[2 * kH + jc[g]];
    bhn[g] = b_hh[2 * kH + jc[g]];
    wiR[g] = w_ih + jc[g] * kF;
    wiZ[g] = w_ih + (kH + jc[g]) * kF;
    wiN[g] = w_ih + (2 * kH + jc[g]) * kF;
    whR[g] = w_hh + jc[g] * kH;
    whZ[g] = w_hh + (kH + jc[g]) * kH;
    whN[g] = w_hh + (2 * kH + jc[g]) * kH;
    fwb[g] = fc1_w + (size_t)jc[g] * kK1;
  }

  v8f accF[2] = {};   // persistent FC1 accumulators (live across the whole t-loop)

  for (int t = 0; t < kT; ++t) {
    v8f aR[2] = {}, aZ[2] = {}, aNi[2] = {}, aNh[2] = {};

    // ---- input-side gates: gi = x_t(16x39) @ w_ih^T, K padded 39->40 ----
    const float* xrow = x + (b0 + n16) * kTF + t * kF;
#pragma unroll
    for (int kk = 0; kk < 10; ++kk) {
      const int kb  = 4 * kk + 2 * hi;      // kb <= 38, always valid
      const int k1  = kb + 1;               // k1 == 39 only at kk=9, hi=1
      const int k1c = (k1 < kF) ? k1 : 0;   // clamped, unconditional loads
      const float m1 = (k1 < kF) ? 1.0f : 0.0f;
      v2f a;
      a.x = xrow[kb];
      a.y = xrow[k1c] * m1;                 // zero A kills the padded k-slice
#pragma unroll
      for (int g = 0; g < 2; ++g) {
        v2f bR, bZ, bN;
        bR.x = wiR[g][kb];  bR.y = wiR[g][k1c];
        bZ.x = wiZ[g][kb];  bZ.y = wiZ[g][k1c];
        bN.x = wiN[g][kb];  bN.y = wiN[g][k1c];
        aR[g]  = wmma_f32_k4(a, bR, aR[g]);
        aZ[g]  = wmma_f32_k4(a, bZ, aZ[g]);
        aNi[g] = wmma_f32_k4(a, bN, aNi[g]);
      }
    }

    // ---- hidden-side gates: gh = h(16x200) @ w_hh^T (K=200, no guards) ----
#pragma unroll 5
    for (int kk = 0; kk < 50; ++kk) {
      const int kb = 4 * kk + 2 * hi;
      const v2f a = *(const v2f*)&hbuf[n16 * kHS + kb];   // 8B-aligned (stride & kb even)
#pragma unroll
      for (int g = 0; g < 2; ++g) {
        const v2f bR = *(const v2f*)(whR[g] + kb);
        const v2f bZ = *(const v2f*)(whZ[g] + kb);
        const v2f bN = *(const v2f*)(whN[g] + kb);
        aR[g]  = wmma_f32_k4(a, bR, aR[g]);
        aZ[g]  = wmma_f32_k4(a, bZ, aZ[g]);
        aNh[g] = wmma_f32_k4(a, bN, aNh[g]);
      }
    }
    __syncthreads();  // all waves done reading h before the gate update rewrites it

    // ---- gates, entirely on accumulator fragments (D layout: row=i+8*hi, col=n16) ----
#pragma unroll
    for (int g = 0; g < 2; ++g) {
      if (vj[g]) {
        const int col = jcol[g];
#pragma unroll
        for (int i = 0; i < 8; ++i) {
          const int m = i + 8 * hi;
          const float r  = sigmoid_f(aR[g][i] + br[g]);
          const float z  = sigmoid_f(aZ[g][i] + bz[g]);
          const float n  = tanh_f(aNi[g][i] + bin[g] + r * (aNh[g][i] + bhn[g]));
          const float hp = hbuf[m * kHS + col];
          hbuf[m * kHS + col] = (1.f - z) * n + z * hp;   // each (m,col) owned by one lane
        }
      }
    }
    __syncthreads();  // h_t visible to all waves

    // ---- FC1 partial: hidden += h_t @ fc1_w[:, t*H:(t+1)*H]^T (K=200, no guards) ----
    const int fo = t * kH;
#pragma unroll 5
    for (int kk = 0; kk < 50; ++kk) {
      const int kb = 4 * kk + 2 * hi;
      const v2f a = *(const v2f*)&hbuf[n16 * kHS + kb];
#pragma unroll
      for (int g = 0; g < 2; ++g) {
        const v2f b = *(const v2f*)(fwb[g] + fo + kb);
        accF[g] = wmma_f32_k4(a, b, accF[g]);
      }
    }
    // no trailing barrier needed: next write to hbuf is after the next
    // __syncthreads() (gate phase of t+1); everything in between only reads.
  }

  // store FC1 accumulators: hidden[b, o]
#pragma unroll
  for (int g = 0; g < 2; ++g) {
    if (vj[g]) {
#pragma unroll
      for (int i = 0; i < 8; ++i) {
        const int m = i + 8 * hi;
        hidden[(b0 + m) * kH + jcol[g]] = accF[g][i];
      }
    }
  }
}

// hidden -> +bias, relu, FC2 (200->10), log_softmax. One wave32 per batch row.
__global__ __launch_bounds__(256)
void fc2_logsoftmax(const float* __restrict__ hidden,
                    const float* __restrict__ fc1_b,
                    const float* __restrict__ fc2_w,
                    const float* __restrict__ fc2_b,
                    float* __restrict__ out) {
  const int wave = threadIdx.x >> 5;
  const int lane = threadIdx.x & 31;
  const int b = blockIdx.x * 8 + wave;
  if (b >= kB) return;

  float p[kC];
#pragma unroll
  for (int c = 0; c < kC; ++c) p[c] = 0.f;

  for (int o = lane; o < kH; o += 32) {
    const float a = fmaxf(hidden[b * kH + o] + fc1_b[o], 0.f);
#pragma unroll
    for (int c = 0; c < kC; ++c) p[c] += a * fc2_w[c * kH + o];
  }
#pragma unroll
  for (int off = 16; off > 0; off >>= 1) {
#pragma unroll
    for (int c = 0; c < kC; ++c) p[c] += __shfl_xor(p[c], off, 32);
  }
  if (lane == 0) {
    float logits[kC], mx = -3.4e38f;
#pragma unroll
    for (int c = 0; c < kC; ++c) { logits[c] = p[c] + fc2_b[c]; mx = fmaxf(mx, logits[c]); }
    float s = 0.f;
#pragma unroll
    for (int c = 0; c < kC; ++c) s += __expf(logits[c] - mx);
    const float lse = mx + __logf(s);
#pragma unroll
    for (int c = 0; c < kC; ++c) out[b * kC + c] = logits[c] - lse;
  }
}

extern "C" void kernel_launch(void* const* d_in, const int* in_sizes, int n_in,
                              void* d_out, int out_size, void* d_ws, size_t ws_size,
                              hipStream_t stream) {
  (void)in_sizes; (void)n_in; (void)out_size; (void)ws_size;
  const float* x     = (const float*)d_in[0];
  const float* h0    = (const float*)d_in[1];
  const float* w_ih  = (const float*)d_in[2];
  const float* w_hh  = (const float*)d_in[3];
  const float* b_ih  = (const float*)d_in[4];
  const float* b_hh  = (const float*)d_in[5];
  const float* fc1_w = (const float*)d_in[6];
  const float* fc1_b = (const float*)d_in[7];
  const float* fc2_w = (const float*)d_in[8];
  const float* fc2_b = (const float*)d_in[9];
  float* out    = (float*)d_out;
  float* hidden = (float*)d_ws;   // 1024 * 200 f32 = 800 KB scratch

  gru_fc1_fused<<<kB / 16, 256, 0, stream>>>(x, h0, w_ih, w_hh, b_ih, b_hh, fc1_w, hidden);
  fc2_logsoftmax<<<kB / 8, 256, 0, stream>>>(hidden, fc1_b, fc2_w, fc2_b, out);
}